// SINSentAddEmb_52295521796615
// MI455X (gfx1250) — compile-verified
//
#include <hip/hip_runtime.h>

typedef float v4f __attribute__((ext_vector_type(4)));
typedef int   v4i __attribute__((ext_vector_type(4)));

#define AS1 __attribute__((address_space(1)))
#define AS3 __attribute__((address_space(3)))

namespace {
constexpr int Bc = 8;
constexpr int Nc = 4096;
constexpr int Hc = 1024;
constexpr float EPSf = 1e-12f;
}

// Grid: Nc blocks of 256 threads (8 wave32s). Wave w handles batch b=w, row n=blockIdx.x.
// LDS holds pe[n] (shared by all 8 batches) + gamma + beta, staged via CDNA5
// async global->LDS loads.
__global__ __launch_bounds__(256) void sinsent_addemb_ln_kernel(
    const int* __restrict__ ssv,      // (B, N, 2) int32
    const float* __restrict__ pe,     // (5000, H)
    const float* __restrict__ gamma,  // (H)
    const float* __restrict__ beta,   // (H)
    float* __restrict__ out)          // (B, N, H)
{
    __shared__ __align__(16) float s_pos[Hc];
    __shared__ __align__(16) float s_g[Hc];
    __shared__ __align__(16) float s_b[Hc];

    const int n    = blockIdx.x;
    const int t    = threadIdx.x;   // 0..255
    const int lane = t & 31;
    const int b    = t >> 5;        // wave index == batch index

    const float* pe_n = pe + (size_t)n * Hc;

    // ---- Stage pe[n], gamma, beta into LDS (4 KB each; 256 thr x 16 B) ----
#if __has_builtin(__builtin_amdgcn_global_load_async_to_lds_b128)
    __builtin_amdgcn_global_load_async_to_lds_b128(
        (AS1 v4i*)(pe_n + t * 4),  (AS3 v4i*)(&s_pos[t * 4]), 0, 0);
    __builtin_amdgcn_global_load_async_to_lds_b128(
        (AS1 v4i*)(gamma + t * 4), (AS3 v4i*)(&s_g[t * 4]),   0, 0);
    __builtin_amdgcn_global_load_async_to_lds_b128(
        (AS1 v4i*)(beta + t * 4),  (AS3 v4i*)(&s_b[t * 4]),   0, 0);
  #if __has_builtin(__builtin_amdgcn_s_wait_asynccnt)
    __builtin_amdgcn_s_wait_asynccnt(0);
  #else
    asm volatile("s_wait_asynccnt 0" ::: "memory");
  #endif
#else
    *(v4f*)&s_pos[t * 4] = *(const v4f*)(pe_n + t * 4);
    *(v4f*)&s_g[t * 4]   = *(const v4f*)(gamma + t * 4);
    *(v4f*)&s_b[t * 4]   = *(const v4f*)(beta + t * 4);
#endif
    __syncthreads();

    // ---- Wave-uniform gather indices for this (b, n) row ----
    const int base = (b * Nc + n) * 2;
    const int para = ssv[base + 0];
    const int sent = ssv[base + 1];
    const float* __restrict__ pa = pe + (size_t)para * Hc;
    const float* __restrict__ pb = pe + (size_t)sent * Hc;

    // ---- Sum three embedding rows; 32 f32 per lane kept in registers ----
    v4f x[8];
    float sum = 0.f, sumsq = 0.f;
#pragma unroll
    for (int i = 0; i < 8; ++i) {
        const int c0 = (i * 32 + lane) * 4;   // 16B-aligned, lane-coalesced
        v4f v = *(const v4f*)&s_pos[c0];
        v += *(const v4f*)(pa + c0);
        v += *(const v4f*)(pb + c0);
        x[i] = v;
        sum   += v.x + v.y + v.z + v.w;
        sumsq += v.x * v.x + v.y * v.y + v.z * v.z + v.w * v.w;
    }

    // ---- wave32 butterfly reduction (mean / variance over H=1024) ----
#pragma unroll
    for (int off = 16; off > 0; off >>= 1) {
        sum   += __shfl_xor(sum,   off, 32);
        sumsq += __shfl_xor(sumsq, off, 32);
    }
    const float invH = 1.0f / (float)Hc;
    const float mean = sum * invH;
    const float var  = sumsq * invH - mean * mean;   // population variance
    const float rstd = rsqrtf(var + EPSf);

    // ---- Normalize, affine, streaming (non-temporal) b128 stores ----
    float* __restrict__ orow = out + ((size_t)b * Nc + n) * Hc;
#pragma unroll
    for (int i = 0; i < 8; ++i) {
        const int c0 = (i * 32 + lane) * 4;
        v4f g  = *(const v4f*)&s_g[c0];
        v4f be = *(const v4f*)&s_b[c0];
        v4f v  = (x[i] - mean) * rstd * g + be;
        __builtin_nontemporal_store(v, (v4f*)(orow + c0));
    }
}

extern "C" void kernel_launch(void* const* d_in, const int* in_sizes, int n_in,
                              void* d_out, int out_size, void* d_ws, size_t ws_size,
                              hipStream_t stream) {
    // d_in[0] = top_vecs  (UNUSED by the reference's output -> never read)
    const int*   ssv   = (const int*)d_in[1];    // (B, N, 2) int32
    const float* pe    = (const float*)d_in[2];  // (5000, H)
    const float* gamma = (const float*)d_in[3];  // (H)
    const float* beta  = (const float*)d_in[4];  // (H)
    float*       out   = (float*)d_out;          // (B, N, H)

    dim3 grid(Nc, 1, 1);
    dim3 block(256, 1, 1);
    hipLaunchKernelGGL(sinsent_addemb_ln_kernel, grid, block, 0, stream,
                       ssv, pe, gamma, beta, out);
}